// RGBDVideoTower_24060406792955
// MI455X (gfx1250) — compile-verified
//
#include <hip/hip_runtime.h>

// Problem constants (fixed by the reference harness)
#define B_ 8
#define N_ 18432
#define C_ 1024
#define K_ 1024
#define THREADS_ 256   // 256 threads * float4 = 1024 floats = one C row

// ---------------------------------------------------------------------------
// gfx1250 async global->LDS path (ASYNCcnt-tracked DMA), guarded so the
// file still compiles on toolchains without the builtins.
// ---------------------------------------------------------------------------
#if defined(__has_builtin)
#if __has_builtin(__builtin_amdgcn_global_load_async_to_lds_b128) && \
    __has_builtin(__builtin_amdgcn_s_wait_asynccnt)
#define HAVE_ASYNC_LDS 1
#endif
#endif
#ifndef HAVE_ASYNC_LDS
#define HAVE_ASYNC_LDS 0
#endif

#if HAVE_ASYNC_LDS
typedef int v4i __attribute__((vector_size(16)));
typedef __attribute__((address_space(1))) v4i* g_v4i_ptr;  // global
typedef __attribute__((address_space(3))) v4i* l_v4i_ptr;  // LDS

__device__ __forceinline__ void async_copy16(float4* lds_dst, const float4* gsrc) {
  // global_load_async_to_lds_b128: 16B per lane, memory -> LDS, ASYNCcnt++
  __builtin_amdgcn_global_load_async_to_lds_b128(
      (g_v4i_ptr)(gsrc),
      (l_v4i_ptr)(lds_dst),
      /*offset=*/0, /*cpol=*/0);
}
#endif

// ---------------------------------------------------------------------------
// Gather kernel (emitted first so the disasm snippet shows the async loop):
// one block per output row (b,k). Walk the sampled voxel's point chain,
// sum feature rows, divide by count. The 4KB feature-row fetch is a
// double-buffered async global->LDS DMA overlapped with the scalar
// next-pointer chase; s_wait_asynccnt <= 1 keeps one transfer in flight.
// ---------------------------------------------------------------------------
__global__ void __launch_bounds__(THREADS_)
voxel_gather_mean(const float* __restrict__ feats,
                  const int* __restrict__ resample,
                  const int* __restrict__ head,
                  const int* __restrict__ next,
                  float* __restrict__ out) {
  const int bk   = blockIdx.x;        // [0, B_*K_)
  const int b    = bk >> 10;          // K_ == 1024
  const int base = b * N_;
  const int t    = threadIdx.x;

  const int seg = resample[bk];       // uniform per block
  int n = head[base + seg];           // uniform per block (scalar chain walk)

  float4 acc; acc.x = 0.f; acc.y = 0.f; acc.z = 0.f; acc.w = 0.f;
  int cnt = 0;

#if HAVE_ASYNC_LDS
  __shared__ float4 buf[2][THREADS_]; // 8KB double buffer; 16B per thread
  int cur = 0;
  if (n >= 0) {
    const float4* row = (const float4*)(feats + ((size_t)(base + n)) * C_);
    async_copy16(&buf[0][t], row + t);
  }
  while (n >= 0) {
    int nn = next[base + n];                        // pointer chase (uniform)
    nn = __builtin_amdgcn_readfirstlane(nn);
    if (nn >= 0) {
      const float4* row2 = (const float4*)(feats + ((size_t)(base + nn)) * C_);
      async_copy16(&buf[cur ^ 1][t], row2 + t);     // prefetch next row
      __builtin_amdgcn_s_wait_asynccnt(1);          // oldest DMA done
    } else {
      __builtin_amdgcn_s_wait_asynccnt(0);          // drain
    }
    // Each thread reads only the 16B it copied itself -> no barrier needed.
    float4 v = buf[cur][t];
    acc.x += v.x; acc.y += v.y; acc.z += v.z; acc.w += v.w;
    ++cnt;
    cur ^= 1;
    n = nn;
  }
#else
  while (n >= 0) {
    int nn = next[base + n];
    nn = __builtin_amdgcn_readfirstlane(nn);
    const float4* row = (const float4*)(feats + ((size_t)(base + n)) * C_);
    float4 v = row[t];                              // coalesced b128 load
    acc.x += v.x; acc.y += v.y; acc.z += v.z; acc.w += v.w;
    ++cnt;
    n = nn;
  }
#endif

  const float c = (float)(cnt > 0 ? cnt : 1);
  float4 r;
  r.x = acc.x / c; r.y = acc.y / c; r.z = acc.z / c; r.w = acc.w / c;
  ((float4*)(out + (size_t)bk * C_))[t] = r;
}

// ---------------------------------------------------------------------------
// Build per-(batch,voxel) linked lists of point indices.
// head[b*N + seg] = most recent point n in that voxel (or -1)
// next[b*N + n]   = previous point in the same voxel (or -1)
// head is pre-set to -1 via hipMemsetAsync(0xFF).
// ---------------------------------------------------------------------------
__global__ void __launch_bounds__(THREADS_)
voxel_build_chains(const int* __restrict__ p2v,
                   int* __restrict__ head,
                   int* __restrict__ next) {
  const int tid = blockIdx.x * THREADS_ + threadIdx.x;   // over B_*N_
  const int b   = tid / N_;
  const int n   = tid - b * N_;
  const int seg = p2v[tid];
  const int prev = atomicExch(&head[b * N_ + seg], n);
  next[tid] = prev;
}

// ---------------------------------------------------------------------------
// Launch: memset head -> build chains -> gather means.
// Inputs: d_in[0]=video_features f32[B,N,C], d_in[1]=p2v i32[B,N],
//         d_in[2]=resample_idx i32[B,K], d_in[3]=num_segments (unused;
//         segment ids are bounded by N, so we size the map by N).
// ws layout: head int[B*N] | next int[B*N]  (~1.2 MB total)
// ---------------------------------------------------------------------------
extern "C" void kernel_launch(void* const* d_in, const int* in_sizes, int n_in,
                              void* d_out, int out_size, void* d_ws, size_t ws_size,
                              hipStream_t stream) {
  (void)in_sizes; (void)n_in; (void)out_size; (void)ws_size;
  const float* feats    = (const float*)d_in[0];
  const int*   p2v      = (const int*)d_in[1];
  const int*   resample = (const int*)d_in[2];

  int* head = (int*)d_ws;
  int* next = head + (size_t)B_ * N_;

  (void)hipMemsetAsync(head, 0xFF, (size_t)B_ * N_ * sizeof(int), stream);  // head = -1

  voxel_build_chains<<<(B_ * N_) / THREADS_, THREADS_, 0, stream>>>(p2v, head, next);

  voxel_gather_mean<<<B_ * K_, THREADS_, 0, stream>>>(
      feats, resample, head, next, (float*)d_out);
}